// BCModel_31439160607295
// MI455X (gfx1250) — compile-verified
//
#include <hip/hip_runtime.h>
#include <math.h>

typedef float v2f_t __attribute__((ext_vector_type(2)));
typedef float v8f_t __attribute__((ext_vector_type(8)));

namespace {
constexpr int B_  = 512;
constexpr int T_  = 256;
constexpr int E_  = 128;
constexpr int H_  = 64;
constexpr int N4_ = 4 * H_;   // 256 (gate width)
}

__device__ __forceinline__ float sigf(float x) { return 1.0f / (1.0f + __expf(-x)); }

// ---------------------------------------------------------------------------
// Kernel 1: G[t*B + b][0:256] = emb[x[b,t]] @ W_ih + b_lstm
// One wave per 16x16 output tile, K=128 via 32x V_WMMA_F32_16X16X4_F32.
// Tiles are ordered M-major so all 8 waves of a block share the same N-tile
// (same W_ih columns + bias -> WGP$ hits on the strided B-fragment loads).
// ---------------------------------------------------------------------------
__global__ void gates_x_kernel(const int* __restrict__ x,      // [B,T]
                               const float* __restrict__ emb,  // [VOCAB,E]
                               const float* __restrict__ Wih,  // [E,N4]
                               const float* __restrict__ bias, // [N4]
                               float* __restrict__ G)          // [T*B,N4]
{
    const int lane = threadIdx.x & 31;
    const int wib  = threadIdx.x >> 5;
    const int tile = blockIdx.x * (blockDim.x >> 5) + wib;
    const int tilesM = (B_ * T_) / 16;           // 8192
    const int tm = tile % tilesM;
    const int tn = tile / tilesM;                // all waves in a block share tn
    const int m0 = tm * 16;
    const int n0 = tn * 16;
    const int hf  = lane >> 4;                   // 0|1
    const int l16 = lane & 15;

    // A-fragment source row: gathered embedding row for (t,b) of row m0+l16
    const int m = m0 + l16;
    const int t = m / B_;
    const int b = m % B_;
    const int idx = x[b * T_ + t];
    const float* arow = emb + (size_t)idx * E_ + 2 * hf;

    // C init: broadcast bias[n] down all 16 rows of the tile
    const float bn = bias[n0 + l16];
    v8f_t acc;
#pragma unroll
    for (int i = 0; i < 8; ++i) acc[i] = bn;

    const float* wcol = Wih + n0 + l16;
#pragma unroll
    for (int k0 = 0; k0 < E_; k0 += 4) {
        v2f_t a = *(const v2f_t*)(arow + k0);    // A: K = k0+2*hf, k0+2*hf+1
        v2f_t bm;
        const int kb = k0 + 2 * hf;
        bm.x = wcol[(size_t)(kb    ) * N4_];     // B: VGPR0 = row kb
        bm.y = wcol[(size_t)(kb + 1) * N4_];     // B: VGPR1 = row kb+1
        acc = __builtin_amdgcn_wmma_f32_16x16x4_f32(
                  false, a, false, bm, (short)0, acc, false, false);
    }

    // D layout: VGPR v, lane -> row m0 + v + 8*hf, col n0 + l16
    float* gout = G + (size_t)m0 * N4_ + n0 + l16;
#pragma unroll
    for (int v = 0; v < 8; ++v)
        gout[(size_t)(v + 8 * hf) * N4_] = acc[v];
}

// ---------------------------------------------------------------------------
// Kernel 2: sequential LSTM scan. Batch rows are independent => each
// workgroup (128 thr = 4 waves) owns 16 batch rows for all T steps.
// Wave w's 4 N-tiles are n0 = 64*j + 16*w  (tile j = gate j, cols [16w,16w+16)),
// so every lane holds i/f/g/o for its own (row,col) elements in-register:
// the LSTM cell update needs NO LDS gate exchange. Only h goes through LDS
// (ping-pong buffer -> a single barrier per step). W_hh fragments live in
// registers; G preactivations are register double-buffered (loads for t+1
// issue during step t's WMMA/transcendental work), with a deep
// global_prefetch_b8 warming G[t+8].
// ---------------------------------------------------------------------------
__global__ void __launch_bounds__(128, 1)
lstm_scan_kernel(const float* __restrict__ G,    // [T*B, N4]
                 const float* __restrict__ Whh,  // [H, N4]
                 const float* __restrict__ h0,   // [B, H]
                 const float* __restrict__ c0,   // [B, H]
                 float* __restrict__ pool)       // [B, 2H] = avg || max
{
    constexpr int HS = 66;   // h row stride: bank-conflict free, 8B aligned
    __shared__ float h_lds[2][16 * HS];

    const int lane = threadIdx.x & 31;
    const int wave = threadIdx.x >> 5;
    const int b0   = blockIdx.x * 16;
    const int hf   = lane >> 4;
    const int l16  = lane & 15;
    const int col  = wave * 16 + l16;    // this lane's hidden column (0..63)
    const int row0 = 8 * hf;             // rows row0..row0+7 per acc slot v

    // ---- preload W_hh fragments: tile j covers gate j, cols [16w,16w+16) ----
    v2f_t bfrag[4][16];
#pragma unroll
    for (int j = 0; j < 4; ++j) {
        const float* wc = Whh + j * 64 + col;
#pragma unroll
        for (int k = 0; k < 16; ++k) {
            const int kb = k * 4 + 2 * hf;
            bfrag[j][k].x = wc[(size_t)(kb    ) * N4_];
            bfrag[j][k].y = wc[(size_t)(kb + 1) * N4_];
        }
    }

    // ---- per-lane cell state / pools, aligned with the D-matrix layout ----
    float c[8], s[8], mx[8];
#pragma unroll
    for (int v = 0; v < 8; ++v) {
        const int rr = row0 + v;
        c[v]  = c0[(size_t)(b0 + rr) * H_ + col];
        s[v]  = 0.0f;
        mx[v] = -1e30f;
        h_lds[0][rr * HS + col] = h0[(size_t)(b0 + rr) * H_ + col];
    }
    __syncthreads();

    // ---- register prefetch of G tile for t = 0 ----
    v8f_t pg[4];
    {
        const float* g_t = G + (size_t)b0 * N4_;
#pragma unroll
        for (int j = 0; j < 4; ++j) {
            const int n = j * 64 + col;
#pragma unroll
            for (int v = 0; v < 8; ++v)
                pg[j][v] = g_t[(size_t)(row0 + v) * N4_ + n];
        }
    }

#pragma unroll 1
    for (int t = 0; t < T_; ++t) {
        // consume prefetched preactivations as the WMMA C operand
        v8f_t acc[4];
#pragma unroll
        for (int j = 0; j < 4; ++j) acc[j] = pg[j];

        // issue next step's G loads (independent of this step's compute)
        const int tnx = (t + 1 < T_) ? (t + 1) : (T_ - 1);
        const float* g_n = G + ((size_t)tnx * B_ + b0) * N4_;
#pragma unroll
        for (int j = 0; j < 4; ++j) {
            const int n = j * 64 + col;
#pragma unroll
            for (int v = 0; v < 8; ++v)
                pg[j][v] = g_n[(size_t)(row0 + v) * N4_ + n];
        }
        // deep prefetch toward WGP$/L2 (128 threads x 128B covers the 16KB tile)
        const int tpf = (t + 8 < T_) ? (t + 8) : (T_ - 1);
        __builtin_prefetch(G + ((size_t)tpf * B_ + b0) * N4_ + threadIdx.x * 32, 0, 0);

        // h @ W_hh : 16 K-steps x 4 gate tiles, A-fragment shared across tiles
        const float* hbuf = h_lds[t & 1];
#pragma unroll
        for (int k = 0; k < 16; ++k) {
            v2f_t a = *(const v2f_t*)(&hbuf[l16 * HS + k * 4 + 2 * hf]);
#pragma unroll
            for (int j = 0; j < 4; ++j)
                acc[j] = __builtin_amdgcn_wmma_f32_16x16x4_f32(
                             false, a, false, bfrag[j][k], (short)0, acc[j],
                             false, false);
        }

        // in-register LSTM cell: acc[0]=i, acc[1]=f, acc[2]=g, acc[3]=o
        float* hnext = h_lds[(t + 1) & 1];
#pragma unroll
        for (int v = 0; v < 8; ++v) {
            const float cn = sigf(acc[1][v]) * c[v] + sigf(acc[0][v]) * tanhf(acc[2][v]);
            const float hn = sigf(acc[3][v]) * tanhf(cn);
            c[v]  = cn;
            s[v]  += hn;
            mx[v] = fmaxf(mx[v], hn);
            hnext[(row0 + v) * HS + col] = hn;   // ping-pong: no WAR on hbuf
        }
        __syncthreads();                          // single barrier per step
    }

    // pools: [B, 2H] = avg || max
#pragma unroll
    for (int v = 0; v < 8; ++v) {
        const int rr = row0 + v;
        pool[(size_t)(b0 + rr) * (2 * H_) + col]      = s[v] * (1.0f / T_);
        pool[(size_t)(b0 + rr) * (2 * H_) + H_ + col] = mx[v];
    }
}

// ---------------------------------------------------------------------------
// Kernel 3: out[b] = sigmoid((pool[b] @ W1 + b1) @ W2 + b2)
// One 64-thread block per batch row (no activation between W1 and W2).
// ---------------------------------------------------------------------------
__global__ void head_kernel(const float* __restrict__ pool, // [B, 2H]
                            const float* __restrict__ W1,   // [2H, 64]
                            const float* __restrict__ b1,   // [64]
                            const float* __restrict__ W2,   // [64, 1]
                            const float* __restrict__ b2,   // [1]
                            float* __restrict__ out)        // [B, 1]
{
    __shared__ float red[64];
    const int b = blockIdx.x;
    const int j = threadIdx.x;
    const float* p = pool + (size_t)b * (2 * H_);
    float acc = b1[j];
#pragma unroll 4
    for (int k = 0; k < 2 * H_; ++k) acc += p[k] * W1[k * 64 + j];
    red[j] = acc * W2[j];
    __syncthreads();
    for (int st = 32; st > 0; st >>= 1) {
        if (j < st) red[j] += red[j + st];
        __syncthreads();
    }
    if (j == 0) out[b] = sigf(red[0] + b2[0]);
}

// ---------------------------------------------------------------------------
extern "C" void kernel_launch(void* const* d_in, const int* in_sizes, int n_in,
                              void* d_out, int out_size, void* d_ws, size_t ws_size,
                              hipStream_t stream) {
    const int*   x      = (const int*)  d_in[0];
    const float* h0     = (const float*)d_in[1];
    const float* c0     = (const float*)d_in[2];
    const float* emb    = (const float*)d_in[3];
    const float* W_ih   = (const float*)d_in[4];
    const float* W_hh   = (const float*)d_in[5];
    const float* b_lstm = (const float*)d_in[6];
    const float* W1     = (const float*)d_in[7];
    const float* b1     = (const float*)d_in[8];
    const float* W2     = (const float*)d_in[9];
    const float* b2     = (const float*)d_in[10];
    float*       out    = (float*)d_out;

    char* ws = (char*)d_ws;
    float* G    = (float*)ws;                                    // [T*B, 256] = 128 MB
    float* pool = (float*)(ws + (size_t)T_ * B_ * N4_ * sizeof(float)); // [B, 128]

    // Kernel 1: gather + x-side GEMM. 131072 tiles, 8 waves per 256-thr block.
    {
        const int tiles = (B_ * T_ / 16) * (N4_ / 16);
        const int wavesPerBlock = 8;
        gates_x_kernel<<<tiles / wavesPerBlock, 32 * wavesPerBlock, 0, stream>>>(
            x, emb, W_ih, b_lstm, G);
    }

    // Kernel 2: per-16-batch-row sequential scan (no inter-WG sync needed).
    lstm_scan_kernel<<<B_ / 16, 128, 0, stream>>>(G, W_hh, h0, c0, pool);

    // Kernel 3: classifier head.
    head_kernel<<<B_, 64, 0, stream>>>(pool, W1, b1, W2, b2, out);
}